// GNN_Graph_14422500180454
// MI455X (gfx1250) — compile-verified
//
#include <hip/hip_runtime.h>
#include <hip/hip_bf16.h>

#define NN  40000
#define EE  400000
#define ESL (EE + NN)
#define DD  64
#define HH  4
#define GG  512

typedef __attribute__((ext_vector_type(2))) float    v2f;
typedef __attribute__((ext_vector_type(8))) float    v8f;
typedef __attribute__((ext_vector_type(4))) unsigned v4u;
typedef __attribute__((ext_vector_type(8))) int      v8i;
typedef __attribute__((ext_vector_type(4))) int      v4i;

__device__ __forceinline__ unsigned fenc(float f) {
    unsigned u = __float_as_uint(f);
    return (u & 0x80000000u) ? ~u : (u | 0x80000000u);
}
__device__ __forceinline__ float fdec(unsigned u) {
    unsigned v = (u & 0x80000000u) ? (u & 0x7FFFFFFFu) : ~u;
    return __uint_as_float(v);
}
#define ENC_NEGMAX 0x00800000u   // fenc(-FLT_MAX)

// ---------------- generic fill / copy ----------------
__global__ void k_fill(unsigned* __restrict__ p, unsigned v, int n) {
    int i = blockIdx.x * 256 + threadIdx.x;
    if (i < n) p[i] = v;
}
__global__ void k_copy(const float* __restrict__ s, float* __restrict__ d, int n) {
    int i = blockIdx.x * 256 + threadIdx.x;
    if (i < n) d[i] = s[i];
}

// ---------------- WMMA f32 GEMM: C[M,N] = A[M,K]@B[K,N] + bias ----------------
// one wave per 16x64 output strip (4 WMMA accumulators share the A fragment).
// M%16==0, N%64==0, K%4==0 (guaranteed by problem dims: M in {40000,512},
// N in {256,192,64}, K in {64,256}).
__global__ void k_gemm_wmma(const float* __restrict__ A, const float* __restrict__ B,
                            const float* __restrict__ bias, float* __restrict__ C,
                            int M, int K, int Nc) {
    int gw = (blockIdx.x * blockDim.x + threadIdx.x) >> 5;
    int tiles_n = Nc >> 6;                   // 64-wide strips
    int tiles = (M >> 4) * tiles_n;
    if (gw >= tiles) return;                 // wave-uniform -> EXEC all-1s for WMMA
    int tm = gw / tiles_n, tn = gw % tiles_n;
    int lane = threadIdx.x & 31;
    int lm = lane & 15;
    int hi = lane >> 4;

    const float* arow = A + (size_t)(tm * 16 + lm) * K;  // A(m,k): lane=m+16*(k>=2), vgpr=k&1
    const float* bstrip = B + tn * 64 + lm;              // B(k,n): lane=n+16*(k>=2), vgpr=k&1
    v8f c0 = {0.f,0.f,0.f,0.f,0.f,0.f,0.f,0.f};
    v8f c1 = c0, c2 = c0, c3 = c0;
    for (int k = 0; k < K; k += 4) {
        v2f a, b;
        a[0] = arow[k + 2 * hi];
        a[1] = arow[k + 2 * hi + 1];
        const float* br0 = bstrip + (size_t)(k + 2 * hi) * Nc;
        const float* br1 = bstrip + (size_t)(k + 2 * hi + 1) * Nc;
        b[0] = br0[0];  b[1] = br1[0];
        c0 = __builtin_amdgcn_wmma_f32_16x16x4_f32(false, a, false, b, (short)0, c0, false, false);
        b[0] = br0[16]; b[1] = br1[16];
        c1 = __builtin_amdgcn_wmma_f32_16x16x4_f32(false, a, false, b, (short)0, c1, false, false);
        b[0] = br0[32]; b[1] = br1[32];
        c2 = __builtin_amdgcn_wmma_f32_16x16x4_f32(false, a, false, b, (short)0, c2, false, false);
        b[0] = br0[48]; b[1] = br1[48];
        c3 = __builtin_amdgcn_wmma_f32_16x16x4_f32(false, a, false, b, (short)0, c3, false, false);
    }
    // C/D layout: vgpr r -> row tm*16 + r + 8*hi, col = strip base + lm
    float* crow = C + (size_t)(tm * 16 + 8 * hi) * Nc + tn * 64 + lm;
    float b0 = bias ? bias[tn * 64 + lm]      : 0.f;
    float b1 = bias ? bias[tn * 64 + 16 + lm] : 0.f;
    float b2 = bias ? bias[tn * 64 + 32 + lm] : 0.f;
    float b3 = bias ? bias[tn * 64 + 48 + lm] : 0.f;
    for (int r = 0; r < 8; ++r) {
        float* cr = crow + (size_t)r * Nc;
        cr[0]  = c0[r] + b0;
        cr[16] = c1[r] + b1;
        cr[32] = c2[r] + b2;
        cr[48] = c3[r] + b3;
    }
}

// ---------------- edge_attr column mean (16 cols) ----------------
__global__ void k_col_mean(const float* __restrict__ ea, float* __restrict__ out) {
    int c = blockIdx.x;                       // 16 blocks
    __shared__ float s[256];
    float acc = 0.f;
    for (int e = threadIdx.x; e < EE; e += 256) acc += ea[(size_t)e * 16 + c];
    s[threadIdx.x] = acc;
    __syncthreads();
    for (int st = 128; st > 0; st >>= 1) {
        if (threadIdx.x < st) s[threadIdx.x] += s[threadIdx.x + st];
        __syncthreads();
    }
    if (threadIdx.x == 0) out[c] = s[0] / (float)EE;
}

// ---------------- GATv2 edge logits: one wave per (edge incl. self-loop) ----------------
// logit[e][h] = sum_d leaky02(xl[src][h*64+d] + xr[dst][h*64+d] + (ea[e]@We)[h*64+d]) * att[h][d]
// We (16x256 = 16KB) is staged into LDS by the Tensor Data Mover (one
// tensor_load_to_lds per block, 1-D tile of 4096 f32), tracked by TENSORcnt.
__global__ void k_edge_logits(const float* __restrict__ xl, const float* __restrict__ xr,
                              const float* __restrict__ ea, const float* __restrict__ eamean,
                              const float* __restrict__ We, const float* __restrict__ att,
                              const int* __restrict__ src, const int* __restrict__ dst,
                              float* __restrict__ logit) {
    __shared__ float sWe[16 * 256];
    __shared__ float sAtt[256];
#if __has_builtin(__builtin_amdgcn_tensor_load_to_lds) && __has_builtin(__builtin_amdgcn_s_wait_tensorcnt)
    if (threadIdx.x < 32) {                       // wave 0 issues the TDM transfer
        unsigned lds = (unsigned)(size_t)(void*)sWe;              // LDS aperture: low 32b = LDS byte offset
        unsigned long long ga = (unsigned long long)(const void*)We;
        v4u g0; v8i g1; v4i g2 = {0, 0, 0, 0}, g3 = {0, 0, 0, 0};
        v8i gx = {0, 0, 0, 0, 0, 0, 0, 0};
        // D# group0: count=1 | lds_addr | global_addr[56:0] | type=2
        g0[0] = 1u;
        g0[1] = lds;
        g0[2] = (unsigned)ga;
        g0[3] = (unsigned)((ga >> 32) & 0x01FFFFFFull) | (2u << 30);
        // D# group1: data_size=4B(code 2); tensor_dim0=tile_dim0=4096; dims1/2 unused
        g1[0] = (int)(2u << 16);                  // wg_mask=0, data_size=2, no flags
        g1[1] = (int)(4096u << 16);               // tensor_dim0[15:0] in high half
        g1[2] = (int)(1u << 16);                  // tensor_dim0[31:16]=0, tensor_dim1=1
        g1[3] = (int)(4096u << 16);               // tensor_dim1[31:16]=0, tile_dim0=4096
        g1[4] = 0;                                // tile_dim1=0, tile_dim2=0 (unused)
        g1[5] = 4096;                             // tensor_dim0_stride[31:0]
        g1[6] = 0;                                // stride hi / dim1_stride lo
        g1[7] = 0;                                // dim1_stride hi
        __builtin_amdgcn_tensor_load_to_lds(g0, g1, g2, g3, gx, 0);
        __builtin_amdgcn_s_wait_tensorcnt((short)0);
    }
#else
    for (int i = threadIdx.x; i < 16 * 256; i += blockDim.x) sWe[i] = We[i];
#endif
    for (int i = threadIdx.x; i < 256; i += blockDim.x) sAtt[i] = att[i];
    __syncthreads();
    int wid = (blockIdx.x * blockDim.x + threadIdx.x) >> 5;
    if (wid >= ESL) return;
    int lane = threadIdx.x & 31;
    int s, d;
    float eav[16];
    if (wid < EE) {
        s = src[wid]; d = dst[wid];
        for (int k = 0; k < 16; ++k) eav[k] = ea[(size_t)wid * 16 + k];
    } else {
        s = d = wid - EE;
        for (int k = 0; k < 16; ++k) eav[k] = eamean[k];
    }
    int base = lane * 8;                       // 8 channels/lane, all in head base/64
    const float* xls = xl + (size_t)s * 256 + base;
    const float* xrd = xr + (size_t)d * 256 + base;
    float part = 0.f;
    for (int j = 0; j < 8; ++j) {
        int c = base + j;
        float ee = 0.f;
        for (int k = 0; k < 16; ++k) ee += eav[k] * sWe[k * 256 + c];
        float m = xls[j] + xrd[j] + ee;
        m = m > 0.f ? m : 0.2f * m;
        part += m * sAtt[c];
    }
    part += __shfl_xor(part, 1, 32);
    part += __shfl_xor(part, 2, 32);
    part += __shfl_xor(part, 4, 32);
    if ((lane & 7) == 0) logit[(size_t)wid * HH + (lane >> 3)] = part;
}

__global__ void k_seg_max(const float* __restrict__ logit, const int* __restrict__ dst,
                          unsigned* __restrict__ mx) {
    int i = blockIdx.x * 256 + threadIdx.x;
    if (i >= ESL * HH) return;
    int e = i >> 2, h = i & 3;
    int d = (e < EE) ? dst[e] : (e - EE);
    atomicMax(&mx[d * HH + h], fenc(logit[i]));
}

__global__ void k_exp_den(const float* __restrict__ logit, const int* __restrict__ dst,
                          const unsigned* __restrict__ mx, float* __restrict__ ex,
                          float* __restrict__ den) {
    int i = blockIdx.x * 256 + threadIdx.x;
    if (i >= ESL * HH) return;
    int e = i >> 2, h = i & 3;
    int d = (e < EE) ? dst[e] : (e - EE);
    float v = expf(logit[i] - fdec(mx[d * HH + h]));
    ex[i] = v;
    atomicAdd(&den[d * HH + h], v);
}

__global__ void k_scatter(const float* __restrict__ xl, const float* __restrict__ ex,
                          const float* __restrict__ den, const int* __restrict__ src,
                          const int* __restrict__ dst, float* __restrict__ acc) {
    int wid = (blockIdx.x * blockDim.x + threadIdx.x) >> 5;
    if (wid >= ESL) return;
    int lane = threadIdx.x & 31;
    int s, d;
    if (wid < EE) { s = src[wid]; d = dst[wid]; }
    else          { s = d = wid - EE; }
    int h = lane >> 3;
    float a = ex[(size_t)wid * HH + h] / (den[d * HH + h] + 1e-16f);
    int base = lane * 8;
    const float* xs = xl + (size_t)s * 256 + base;
    float* ad = acc + (size_t)d * 256 + base;
    for (int j = 0; j < 8; ++j) atomicAdd(&ad[j], a * xs[j]);
}

// ---------------- BN + ELU ----------------
__global__ void k_bn_elu(const float* __restrict__ acc, const float* __restrict__ bias,
                         const float* __restrict__ gamma, const float* __restrict__ beta,
                         const float* __restrict__ mean, const float* __restrict__ var,
                         float* __restrict__ h, int count, int hd) {
    int i = blockIdx.x * 256 + threadIdx.x;
    if (i >= count) return;
    int c = i % hd;
    float v = acc[i] + bias[c];
    v = (v - mean[c]) * rsqrtf(var[c] + 1e-5f) * gamma[c] + beta[c];
    h[i] = v > 0.f ? v : expm1f(v);
}

__global__ void k_bn_elu_mean(const float* __restrict__ acc, const float* __restrict__ bias,
                              const float* __restrict__ gamma, const float* __restrict__ beta,
                              const float* __restrict__ mean, const float* __restrict__ var,
                              float* __restrict__ h) {
    int i = blockIdx.x * 256 + threadIdx.x;
    if (i >= NN * DD) return;
    int n = i >> 6, c = i & 63;
    const float* a = acc + (size_t)n * 256;
    float v = 0.25f * (a[c] + a[64 + c] + a[128 + c] + a[192 + c]) + bias[c];
    v = (v - mean[c]) * rsqrtf(var[c] + 1e-5f) * gamma[c] + beta[c];
    h[i] = v > 0.f ? v : expm1f(v);
}

// ---------------- GRU combine (+leaky) over D=64, gates stride 192 ----------------
__global__ void k_gru(const float* __restrict__ gi, const float* __restrict__ gh,
                      float* __restrict__ x, int rows, float slope) {
    int i = blockIdx.x * 256 + threadIdx.x;
    if (i >= rows * DD) return;
    int n = i >> 6, c = i & 63;
    const float* gin = gi + (size_t)n * 192;
    const float* ghn = gh + (size_t)n * 192;
    float r = 1.f / (1.f + expf(-(gin[c] + ghn[c])));
    float z = 1.f / (1.f + expf(-(gin[64 + c] + ghn[64 + c])));
    float nn = tanhf(gin[128 + c] + r * ghn[128 + c]);
    float v = (1.f - z) * nn + z * x[i];
    x[i] = v > 0.f ? v : slope * v;
}

// ---------------- graph pooling ----------------
__global__ void k_pool(const float* __restrict__ x, const int* __restrict__ batch,
                       float* __restrict__ og) {
    int i = blockIdx.x * 256 + threadIdx.x;
    if (i >= NN * DD) return;
    int n = i >> 6, c = i & 63;
    atomicAdd(&og[(size_t)batch[n] * DD + c], x[i]);
}
__global__ void k_leaky(float* __restrict__ p, int n, float slope) {
    int i = blockIdx.x * 256 + threadIdx.x;
    if (i >= n) return;
    float v = p[i];
    p[i] = v > 0.f ? v : slope * v;
}

// ---------------- mol attention (heads=1, bipartite node->graph) ----------------
__global__ void k_mol_logits(const float* __restrict__ xlm, const float* __restrict__ xrm,
                             const float* __restrict__ att, const int* __restrict__ batch,
                             float* __restrict__ lg) {
    int wid = (blockIdx.x * blockDim.x + threadIdx.x) >> 5;
    if (wid >= NN) return;
    int lane = threadIdx.x & 31;
    int b = batch[wid];
    float p = 0.f;
    for (int c = lane; c < DD; c += 32) {
        float m = xlm[(size_t)wid * DD + c] + xrm[(size_t)b * DD + c];
        m = m > 0.f ? m : 0.2f * m;
        p += m * att[c];
    }
    for (int o = 16; o > 0; o >>= 1) p += __shfl_xor(p, o, 32);
    if (lane == 0) lg[wid] = p;
}
__global__ void k_mol_max(const float* __restrict__ lg, const int* __restrict__ batch,
                          unsigned* __restrict__ mx) {
    int i = blockIdx.x * 256 + threadIdx.x;
    if (i >= NN) return;
    atomicMax(&mx[batch[i]], fenc(lg[i]));
}
__global__ void k_mol_expden(const float* __restrict__ lg, const int* __restrict__ batch,
                             const unsigned* __restrict__ mx, float* __restrict__ ex,
                             float* __restrict__ den) {
    int i = blockIdx.x * 256 + threadIdx.x;
    if (i >= NN) return;
    float v = expf(lg[i] - fdec(mx[batch[i]]));
    ex[i] = v;
    atomicAdd(&den[batch[i]], v);
}
__global__ void k_mol_scatter(const float* __restrict__ xlm, const float* __restrict__ ex,
                              const float* __restrict__ den, const int* __restrict__ batch,
                              float* __restrict__ hg, float* __restrict__ aacc) {
    int wid = (blockIdx.x * blockDim.x + threadIdx.x) >> 5;
    if (wid >= NN) return;
    int lane = threadIdx.x & 31;
    int b = batch[wid];
    float a = ex[wid] / (den[b] + 1e-16f);
    if (lane == 0) aacc[wid] += a;
    for (int c = lane; c < DD; c += 32)
        atomicAdd(&hg[(size_t)b * DD + c], a * xlm[(size_t)wid * DD + c]);
}

// ---------------- final linear + att outputs ----------------
__global__ void k_lin(const float* __restrict__ og, const float* __restrict__ W,
                      const float* __restrict__ b, float* __restrict__ out) {
    int g = blockIdx.x * 256 + threadIdx.x;
    if (g >= GG) return;
    float s = b[0];
    for (int c = 0; c < DD; ++c) s += og[(size_t)g * DD + c] * W[c];
    out[g] = s;
}
__global__ void k_write_att(const int* __restrict__ batch, const float* __restrict__ aacc,
                            float* __restrict__ out) {
    int i = blockIdx.x * 256 + threadIdx.x;
    if (i >= NN) return;
    out[GG + i] = (float)i;
    out[GG + NN + i] = (float)batch[i];
    out[GG + 2 * NN + i] = 0.5f * aacc[i];
}

// ======================================================================
static inline int gblk(long long t) { return (int)((t + 255) / 256); }

extern "C" void kernel_launch(void* const* d_in, const int* in_sizes, int n_in,
                              void* d_out, int out_size, void* d_ws, size_t ws_size,
                              hipStream_t stream) {
    const float* x_in = (const float*)d_in[0];
    const int* ei     = (const int*)d_in[1];
    const float* ea   = (const float*)d_in[2];
    const int* batch  = (const int*)d_in[3];
    const int* src = ei;
    const int* dst = ei + EE;

    auto P = [&](int i) { return (const float*)d_in[i]; };

    // workspace layout (float units)
    float* W = (float*)d_ws;
    size_t o = 0;
    float* XL  = W + o; o += (size_t)NN * 256;
    float* XR  = W + o; o += (size_t)NN * 256;   // doubles as H after logits done
    float* ACC = W + o; o += (size_t)NN * 256;
    float* GI  = W + o; o += (size_t)NN * 192;
    float* GH  = W + o; o += (size_t)NN * 192;
    float* LG  = W + o; o += (size_t)ESL * HH;
    float* EX  = W + o; o += (size_t)ESL * HH;
    unsigned* MX  = (unsigned*)(W + o); o += (size_t)NN * HH;
    float* DEN = W + o; o += (size_t)NN * HH;
    float* XC  = W + o; o += (size_t)NN * DD;
    float* EAM = W + o; o += 16;
    float* OG  = W + o; o += (size_t)GG * DD;
    float* HG  = W + o; o += (size_t)GG * DD;
    float* GIG = W + o; o += (size_t)GG * 192;
    float* GHG = W + o; o += (size_t)GG * 192;
    float* XRG = W + o; o += (size_t)GG * DD;
    float* MLG = W + o; o += NN;
    float* MEX = W + o; o += NN;
    unsigned* MMX = (unsigned*)(W + o); o += GG;
    float* MDEN = W + o; o += GG;
    float* AAC  = W + o; o += NN;
    float* H = XR;   // alias: XR dead once logits computed

    auto gemm = [&](const float* A, const float* B, const float* bias, float* C,
                    int M, int K, int Nc) {
        long long waves = (long long)(M / 16) * (Nc / 64);
        k_gemm_wmma<<<gblk(waves * 32), 256, 0, stream>>>(A, B, bias, C, M, K, Nc);
    };
    auto fillf = [&](float* p, int n) {
        k_fill<<<gblk(n), 256, 0, stream>>>((unsigned*)p, 0u, n);
    };

    k_copy<<<gblk((long long)NN * DD), 256, 0, stream>>>(x_in, XC, NN * DD);
    k_col_mean<<<16, 256, 0, stream>>>(ea, EAM);

    // ---------------- 3 GATv2+GRU layers ----------------
    for (int l = 0; l < 3; ++l) {
        int li = 4 + l * 15;
        const float* bn_beta = P(li + 0), *bn_gamma = P(li + 1);
        const float* bn_mean = P(li + 2), *bn_var = P(li + 3);
        const float* We = P(li + 4), *Wl = P(li + 5), *Wr = P(li + 6);
        const float* att = P(li + 7), *bias = P(li + 8);
        const float* bl = P(li + 9), *br = P(li + 10);
        const float* Whh = P(li + 11), *Wih = P(li + 12);
        const float* bhh = P(li + 13), *bih = P(li + 14);
        int concat = (l < 2);
        int hdim = concat ? 256 : 64;

        k_fill<<<gblk(NN * HH), 256, 0, stream>>>(MX, ENC_NEGMAX, NN * HH);
        fillf(DEN, NN * HH);
        fillf(ACC, NN * 256);

        gemm(XC, Wl, bl, XL, NN, DD, 256);
        gemm(XC, Wr, br, XR, NN, DD, 256);

        k_edge_logits<<<gblk((long long)ESL * 32), 256, 0, stream>>>(
            XL, XR, ea, EAM, We, att, src, dst, LG);
        k_seg_max<<<gblk((long long)ESL * HH), 256, 0, stream>>>(LG, dst, MX);
        k_exp_den<<<gblk((long long)ESL * HH), 256, 0, stream>>>(LG, dst, MX, EX, DEN);
        k_scatter<<<gblk((long long)ESL * 32), 256, 0, stream>>>(XL, EX, DEN, src, dst, ACC);

        if (concat)
            k_bn_elu<<<gblk((long long)NN * 256), 256, 0, stream>>>(
                ACC, bias, bn_gamma, bn_beta, bn_mean, bn_var, H, NN * 256, 256);
        else
            k_bn_elu_mean<<<gblk((long long)NN * DD), 256, 0, stream>>>(
                ACC, bias, bn_gamma, bn_beta, bn_mean, bn_var, H);

        gemm(H, Wih, bih, GI, NN, hdim, 192);
        gemm(XC, Whh, bhh, GH, NN, DD, 192);
        k_gru<<<gblk((long long)NN * DD), 256, 0, stream>>>(GI, GH, XC, NN, 0.01f);
    }

    // ---------------- graph pooling ----------------
    fillf(OG, GG * DD);
    k_pool<<<gblk((long long)NN * DD), 256, 0, stream>>>(XC, batch, OG);
    k_leaky<<<gblk(GG * DD), 256, 0, stream>>>(OG, GG * DD, 0.01f);
    fillf(AAC, NN);

    // ---------------- 2 mol attention layers ----------------
    const float* m_beta = P(51), *m_gamma = P(52), *m_mean = P(53), *m_var = P(54);
    const float* mWl = P(55), *mWr = P(56), *m_att = P(57), *m_bias = P(58);
    const float* m_bl = P(59), *m_br = P(60);
    const float* mWhh = P(61), *mWih = P(62), *m_bhh = P(63), *m_bih = P(64);

    for (int it = 0; it < 2; ++it) {
        gemm(XC, mWl, m_bl, XL, NN, DD, DD);
        gemm(OG, mWr, m_br, XRG, GG, DD, DD);
        k_fill<<<gblk(GG), 256, 0, stream>>>(MMX, ENC_NEGMAX, GG);
        fillf(MDEN, GG);
        fillf(HG, GG * DD);

        k_mol_logits<<<gblk((long long)NN * 32), 256, 0, stream>>>(XL, XRG, m_att, batch, MLG);
        k_mol_max<<<gblk(NN), 256, 0, stream>>>(MLG, batch, MMX);
        k_mol_expden<<<gblk(NN), 256, 0, stream>>>(MLG, batch, MMX, MEX, MDEN);
        k_mol_scatter<<<gblk((long long)NN * 32), 256, 0, stream>>>(XL, MEX, MDEN, batch, HG, AAC);

        k_bn_elu<<<gblk(GG * DD), 256, 0, stream>>>(
            HG, m_bias, m_gamma, m_beta, m_mean, m_var, HG, GG * DD, DD);

        gemm(HG, mWih, m_bih, GIG, GG, DD, 192);
        gemm(OG, mWhh, m_bhh, GHG, GG, DD, 192);
        k_gru<<<gblk(GG * DD), 256, 0, stream>>>(GIG, GHG, OG, GG, 0.01f);
    }

    // ---------------- outputs ----------------
    float* out = (float*)d_out;
    k_lin<<<gblk(GG), 256, 0, stream>>>(OG, P(49), P(50), out);
    k_write_att<<<gblk(NN), 256, 0, stream>>>(batch, AAC, out);
}